// Model_27633819582794
// MI455X (gfx1250) — compile-verified
//
#include <hip/hip_runtime.h>
#include <stdint.h>
#include <math.h>

// ---------------------------------------------------------------------------
// Types
// ---------------------------------------------------------------------------
typedef float v2f __attribute__((ext_vector_type(2)));
typedef float v8f __attribute__((ext_vector_type(8)));
typedef unsigned int u32;
typedef u32 v4u __attribute__((ext_vector_type(4)));
typedef int  v4i __attribute__((ext_vector_type(4)));
typedef int  v8i __attribute__((ext_vector_type(8)));

struct cf { float x, y; };
__device__ __forceinline__ cf cmk(float x, float y){ cf r; r.x = x; r.y = y; return r; }
__device__ __forceinline__ cf cadd(cf a, cf b){ return cmk(a.x + b.x, a.y + b.y); }
__device__ __forceinline__ cf csub(cf a, cf b){ return cmk(a.x - b.x, a.y - b.y); }
__device__ __forceinline__ cf cmul(cf a, cf b){ return cmk(a.x*b.x - a.y*b.y, a.x*b.y + a.y*b.x); }
__device__ __forceinline__ cf cscale(cf a, float s){ return cmk(a.x*s, a.y*s); }
__device__ __forceinline__ cf cneg(cf a){ return cmk(-a.x, -a.y); }
__device__ __forceinline__ cf cdiv(cf a, cf b){
  float inv = 1.0f / (b.x*b.x + b.y*b.y);
  return cmk((a.x*b.x + a.y*b.y)*inv, (a.y*b.x - a.x*b.y)*inv);
}
// principal-branch complex sqrt (matches jnp.sqrt on complex64)
__device__ __forceinline__ cf csqrtc(cf z){
  float m  = sqrtf(z.x*z.x + z.y*z.y);
  float re = sqrtf(fmaxf(0.5f*(m + z.x), 0.0f));
  float im = sqrtf(fmaxf(0.5f*(m - z.x), 0.0f));
  return cmk(re, copysignf(im, z.y));
}

struct m2 { cf a, b, c, d; };  // [[a,b],[c,d]]
__device__ __forceinline__ m2 mmul(const m2& X, const m2& Y){
  m2 R;
  R.a = cadd(cmul(X.a, Y.a), cmul(X.b, Y.c));
  R.b = cadd(cmul(X.a, Y.b), cmul(X.b, Y.d));
  R.c = cadd(cmul(X.c, Y.a), cmul(X.d, Y.c));
  R.d = cadd(cmul(X.c, Y.b), cmul(X.d, Y.d));
  return R;
}
// [[1,r],[r,1]] * it
__device__ __forceinline__ m2 iface_rt(cf r, cf it){
  m2 M; M.a = it; M.b = cmul(it, r); M.c = M.b; M.d = it; return M;
}

// real 4x4 embedding of a complex 2x2: R = [[X,-Y],[Y,X]]
__device__ __forceinline__ void stage_R(float* p, const m2& M){
  p[0]  = M.a.x; p[1]  = M.b.x; p[2]  = -M.a.y; p[3]  = -M.b.y;
  p[4]  = M.c.x; p[5]  = M.d.x; p[6]  = -M.c.y; p[7]  = -M.d.y;
  p[8]  = M.a.y; p[9]  = M.b.y; p[10] =  M.a.x; p[11] =  M.b.x;
  p[12] = M.c.y; p[13] = M.d.y; p[14] =  M.c.x; p[15] =  M.d.x;
}

// ---------------------------------------------------------------------------
// TDM: DMA small f32 array from global into LDS (1-D tensor descriptor)
// 6-arg builtin form (clang-23 / therock-10.0 toolchain)
// ---------------------------------------------------------------------------
__device__ __forceinline__ void tdm_load_f32(const float* gsrc, u32 lds_off, int nelem){
  unsigned long long ga = (unsigned long long)(uintptr_t)gsrc;
  v4u g0;
  g0[0] = 1u;                                             // count=1, user descriptor
  g0[1] = lds_off;                                        // lds_addr (bytes)
  g0[2] = (u32)(ga & 0xffffffffull);                      // global_addr[31:0]
  g0[3] = (u32)((ga >> 32) & 0x1ffffffull) | (2u << 30);  // global_addr[56:32] | type=2
  u32 n16 = (u32)nelem & 0xffffu;
  v8i g1;
  g1[0] = (int)(2u << 16);      // data_size = 4 bytes; workgroup_mask=0 (not in cluster)
  g1[1] = (int)(n16 << 16);     // tensor_dim0 (low 16 bits at [63:48])
  g1[2] = (int)(1u  << 16);     // tensor_dim1 = 1
  g1[3] = (int)(n16 << 16);     // tile_dim0 = nelem
  g1[4] = 1;                    // tile_dim1 = 1
  g1[5] = nelem;                // tensor_dim0_stride (low 32)
  g1[6] = (int)(n16 << 16);     // tensor_dim1_stride (low 16)
  g1[7] = 0;
  v4i g2 = {0,0,0,0};           // higher-dim groups unused (tile_dim2 = 0)
  v4i g3 = {0,0,0,0};
  v8i g4 = {0,0,0,0,0,0,0,0};
  __builtin_amdgcn_tensor_load_to_lds(g0, g1, g2, g3, g4, 0);
}

// ---------------------------------------------------------------------------
// Final T * D via V_WMMA_F32_16X16X4_F32, 4 points per WMMA (real 4x4 embed)
// res[0..3] = real(m00,m01,m10,m11), res[4..7] = imag(m00,m01,m10,m11)
// ---------------------------------------------------------------------------
__device__ __forceinline__ void final_mul_wmma(
    const m2& T, const m2& D, int wave, int lane,
    float (&sA)[8][32][16], float (&sB)[8][32][16], float (&sC)[8][32][8],
    float (&res)[8])
{
  stage_R(sA[wave][lane], T);
  stage_R(sB[wave][lane], D);
  __syncthreads();

  const int h = lane >> 4;     // wave half
  const int n = lane & 15;     // A row / B col index
  const int q = n >> 2;        // 4x4 block index within the 16x16 tile
  const int r = n & 3;         // row/col within the block
  const bool contrib = ((h == 0) ? (q < 2) : (q >= 2)) && (r < 2);
  const int vbase = 4 * (q - 2 * h);   // C VGPR base of this block's rows

#pragma unroll
  for (int g = 0; g < 8; ++g) {        // 8 WMMAs cover the wave's 32 points
    const int pt = g * 4 + q;
    v2f av, bv;
    // A 16x4: lane=M, VGPR v + half h -> K = 2h+v  => A[M][K] = R(T_pt)[r][K]
    av[0] = sA[wave][pt][r * 4 + 2 * h];
    av[1] = sA[wave][pt][r * 4 + 2 * h + 1];
    // B 4x16: lane=N, VGPR v + half h -> K = 2h+v  => B[K][N] = R(D_pt)[K][r]
    bv[0] = sB[wave][pt][(2 * h) * 4 + r];
    bv[1] = sB[wave][pt][(2 * h + 1) * 4 + r];
    v8f cz = {0.f, 0.f, 0.f, 0.f, 0.f, 0.f, 0.f, 0.f};
    v8f c = __builtin_amdgcn_wmma_f32_16x16x4_f32(
        false, av, false, bv, (short)0, cz, false, false);
    if (contrib) {
      const bool lo = (vbase == 0);
      float e0 = lo ? c[0] : c[4];
      float e1 = lo ? c[1] : c[5];
      float e2 = lo ? c[2] : c[6];
      float e3 = lo ? c[3] : c[7];
      sC[wave][pt][0 + r] = e0;   // R[0][r]  -> real row 0
      sC[wave][pt][2 + r] = e1;   // R[1][r]  -> real row 1
      sC[wave][pt][4 + r] = e2;   // R[2][r]  -> imag row 0
      sC[wave][pt][6 + r] = e3;   // R[3][r]  -> imag row 1
    }
  }
  __syncthreads();
#pragma unroll
  for (int k = 0; k < 8; ++k) res[k] = sC[wave][lane][k];
  __syncthreads();  // buffers reused by second polarization
}

// ---------------------------------------------------------------------------
// Kernel: one block per wavelength, one thread per angle
// ---------------------------------------------------------------------------
__global__ __launch_bounds__(256) void tmm_kernel(
    const float* __restrict__ wavelengths,
    const float* __restrict__ angles,
    const float* __restrict__ n_real,
    const float* __restrict__ n_imag,
    const float* __restrict__ thickness,
    const float* __restrict__ n_env,
    const float* __restrict__ n_subs,
    float* __restrict__ out,
    int L, int A, int N)
{
  __shared__ float sNr[64], sNi[64], sTh[64];
  __shared__ float sA[8][32][16];
  __shared__ float sB[8][32][16];
  __shared__ float sC[8][32][8];

  if (threadIdx.x < 32) {  // wave 0 issues the tensor DMAs
    tdm_load_f32(n_real,    (u32)(uintptr_t)(void*)sNr, N);
    tdm_load_f32(n_imag,    (u32)(uintptr_t)(void*)sNi, N);
    tdm_load_f32(thickness, (u32)(uintptr_t)(void*)sTh, N);
    __builtin_amdgcn_s_wait_tensorcnt(0);
  }
  __syncthreads();

  const int l    = blockIdx.x;
  const int a    = threadIdx.x;
  const int wave = a >> 5;
  const int lane = a & 31;

  const float wl = wavelengths[l];
  const float th = angles[a] * 0.017453292519943295f;
  const float sn = __sinf(th);

  const cf ne = cmk(n_env[0], n_env[1]);
  const cf ns = cmk(n_subs[0], n_subs[1]);

  const cf nx  = cscale(ne, sn);
  const cf nx2 = cmul(nx, nx);
  const cf ne2 = cmul(ne, ne);
  const cf ns2 = cmul(ns, ns);

  const cf kzE = csqrtc(csub(ne2, nx2));
  const cf kzA = csqrtc(csub(cmk(1.f, 0.f), nx2));
  const cf kzS = csqrtc(csub(ns2, nx2));

  // T0: env -> air
  cf sumEA = cadd(kzE, kzA);
  m2 Ts = iface_rt(cdiv(csub(kzE, kzA), sumEA),
                   cdiv(sumEA, cscale(kzE, 2.f)));
  cf den0 = cadd(kzE, cmul(ne2, kzA));                 // n2^2 kz1 + n1^2 kz2 (n2=1)
  m2 Tp = iface_rt(cdiv(csub(kzE, cmul(ne2, kzA)), den0),
                   cdiv(den0, cscale(cmul(ne, kzE), 2.f)));

  for (int lyr = 0; lyr < N; ++lyr) {
    const cf nl  = cmk(sNr[lyr], sNi[lyr]);
    const cf nl2 = cmul(nl, nl);
    const cf kz  = csqrtc(csub(nl2, nx2));
    const float cph = 6.283185307179586f * sTh[lyr] / wl;
    const float pr = cph * kz.x;
    const float pi = cph * kz.y;
    const float sp = __sinf(pr), cp = __cosf(pr);
    const float gp = __expf(pi), gm = __expf(-pi);
    m2 P;
    P.a = cmk(gp * cp, -gp * sp);   // exp(-i*phi)
    P.b = cmk(0.f, 0.f);
    P.c = cmk(0.f, 0.f);
    P.d = cmk(gm * cp,  gm * sp);   // exp(+i*phi)

    // s-pol: air -> layer -> air
    {
      cf sum = cadd(kzA, kz);
      cf r   = cdiv(csub(kzA, kz), sum);
      cf it1 = cdiv(sum, cscale(kzA, 2.f));
      cf it2 = cdiv(sum, cscale(kz,  2.f));
      m2 Din  = iface_rt(r, it1);
      m2 Dout = iface_rt(cneg(r), it2);
      Ts = mmul(Ts, mmul(mmul(Din, P), Dout));
    }
    // p-pol
    {
      cf t1  = cmul(nl2, kzA);           // nl^2 * kzA
      cf den = cadd(t1, kz);
      cf r   = cdiv(csub(t1, kz), den);
      cf it1 = cdiv(den, cscale(cmul(nl, kzA), 2.f));
      cf it2 = cdiv(den, cscale(cmul(nl, kz),  2.f));
      m2 Din  = iface_rt(r, it1);
      m2 Dout = iface_rt(cneg(r), it2);
      Tp = mmul(Tp, mmul(mmul(Din, P), Dout));
    }
  }

  // Final air -> substrate interfaces
  cf sumAS = cadd(kzA, kzS);
  m2 Ds = iface_rt(cdiv(csub(kzA, kzS), sumAS),
                   cdiv(sumAS, cscale(kzA, 2.f)));
  cf denF = cadd(cmul(ns2, kzA), kzS);
  m2 Dp = iface_rt(cdiv(csub(cmul(ns2, kzA), kzS), denF),
                   cdiv(denF, cscale(cmul(ns, kzA), 2.f)));

  float rs[8], rp[8];
  final_mul_wmma(Ts, Ds, wave, lane, sA, sB, sC, rs);  // T_s * D_s on matrix core
  final_mul_wmma(Tp, Dp, wave, lane, sA, sB, sC, rp);  // T_p * D_p on matrix core

  // out shape [4, L, A, 2, 2]; write each 2x2 as one 16B store
  float4* o = reinterpret_cast<float4*>(out);
  const size_t p4 = (size_t)l * (size_t)A + (size_t)a;
  const size_t c4 = (size_t)L * (size_t)A;
  o[0 * c4 + p4] = make_float4(rs[0], rs[1], rs[2], rs[3]);  // Ts.re
  o[1 * c4 + p4] = make_float4(rs[4], rs[5], rs[6], rs[7]);  // Ts.im
  o[2 * c4 + p4] = make_float4(rp[0], rp[1], rp[2], rp[3]);  // Tp.re
  o[3 * c4 + p4] = make_float4(rp[4], rp[5], rp[6], rp[7]);  // Tp.im
}

// ---------------------------------------------------------------------------
// Harness entry point
// ---------------------------------------------------------------------------
extern "C" void kernel_launch(void* const* d_in, const int* in_sizes, int n_in,
                              void* d_out, int out_size, void* d_ws, size_t ws_size,
                              hipStream_t stream) {
  const float* wavelengths = (const float*)d_in[0];
  const float* angles      = (const float*)d_in[1];
  const float* n_real      = (const float*)d_in[2];
  const float* n_imag      = (const float*)d_in[3];
  const float* thickness   = (const float*)d_in[4];
  const float* n_env       = (const float*)d_in[5];
  const float* n_subs      = (const float*)d_in[6];
  float* out = (float*)d_out;

  const int L = in_sizes[0];   // 2048 wavelengths -> blocks
  const int A = in_sizes[1];   // 256 angles      -> threads per block
  const int N = in_sizes[2];   // 16 layers

  tmm_kernel<<<dim3(L), dim3(A), 0, stream>>>(
      wavelengths, angles, n_real, n_imag, thickness, n_env, n_subs,
      out, L, A, N);
}